// MetapopLayer_36962488549821
// MI455X (gfx1250) — compile-verified
//
#include <hip/hip_runtime.h>

// CDNA5 wave32 WMMA vector types
typedef __attribute__((ext_vector_type(2))) float v2f;
typedef __attribute__((ext_vector_type(8))) float v8f;

constexpr int Mdim = 64;   // patches
constexpr int Cdim = 4;    // compartments
constexpr int APAD = 68;   // padded LDS row stride (floats): 16 rows * 4-bank step -> all 64 banks
constexpr float CLIP_MAX = 1e10f;

// Raw hardware transcendentals: v_log_f32 is log2, v_exp_f32 is exp2.
// prod_j(1-x) == 2^(sum_j log2(1-x)), so base-2 end-to-end is exact math-wise
// and avoids the ln2 hi/lo FMA pair + denorm/inf fixups of precise logf/expf.
__device__ __forceinline__ float hw_log2(float x) { return __builtin_amdgcn_logf(x); }
__device__ __forceinline__ float hw_exp2(float x) { return __builtin_amdgcn_exp2f(x); }

// One block (128 thr = 4 waves) per ensemble member n. Everything LDS-resident
// across the 100-step scan; WMMA f32 16x16x4 does the two 64-wide row
// reductions (log2-sum via B=ones, matvec via B=broadcast(p)).
__launch_bounds__(128, 1)
__global__ void metapop_step_kernel(const float* __restrict__ R,
                                    const float* __restrict__ T,
                                    const float* __restrict__ rho0,
                                    const float* __restrict__ beta,
                                    float* __restrict__ out,
                                    int N, int timesteps)
{
    __shared__ float A_lds[Mdim * APAD];   // A[i][j] = Rt[n,i,j] * beta/ntot[j]  (loop-invariant)
    __shared__ float Rl[Mdim * APAD];      // R[n], row-major, padded
    __shared__ float rho_lds[Mdim * Cdim];
    __shared__ float g_lds[Mdim];          // beta / ntot
    __shared__ float s_lds[Mdim];
    __shared__ float p_lds[Mdim];
    __shared__ float q_lds[Mdim];

    const int n    = blockIdx.x;
    const int tid  = threadIdx.x;
    const int lid  = tid & 31;        // lane in wave32
    const int wv   = tid >> 5;        // wave id = 16-row tile id
    const int i0   = wv * 16;
    const int hi   = (lid >> 4) & 1;  // 0: K=0..1, 1: K=2..3 (A-operand layout)
    const int arow = i0 + (lid & 15); // this lane's A-matrix row (fixed all step)

    const float* Rn = R + (size_t)n * (Mdim * Mdim);

    // ---- P1: copy R[n] -> LDS (coalesced float4, padded rows) ----
    {
        const int row  = tid >> 1;
        const int half = (tid & 1) * 32;
        const float4* src = (const float4*)(Rn + row * Mdim + half);
        #pragma unroll
        for (int v = 0; v < 8; ++v)
            *(float4*)&Rl[row * APAD + half + v * 4] = src[v];
    }
    __syncthreads();

    // ---- P2: ntot[k] = sum_j R[n,j,k] ;  g[k] = beta / ntot[k] ----
    const float beta_n = beta[n];
    if (tid < Mdim) {
        float sum = 0.f;
        for (int j = 0; j < Mdim; ++j) sum += Rl[j * APAD + tid];
        g_lds[tid] = beta_n / sum;
    }
    __syncthreads();

    // ---- P3: gather A[i][j] = Rt[n,i,j] * g[j] ----
    // Faithful full-reverse-transpose reshape: for flat = n*M*M + i*M + j,
    // Rt element = R[flat % N][(flat/N) % M][flat / (M*N)].
    {
        const unsigned MN = (unsigned)Mdim * (unsigned)N;
        for (int e = tid; e < Mdim * Mdim; e += 128) {
            const unsigned i = (unsigned)e >> 6;
            const unsigned j = (unsigned)e & 63u;
            const unsigned flat = (unsigned)n * (unsigned)(Mdim * Mdim) + (unsigned)e;
            const unsigned a    = flat / MN;
            const unsigned b    = (flat / (unsigned)N) % (unsigned)Mdim;
            const unsigned srcn = flat % (unsigned)N;
            const float rv = R[(size_t)srcn * (Mdim * Mdim) + b * Mdim + a];
            A_lds[i * APAD + j] = rv * g_lds[j];
        }
    }

    // ---- P4: rho0 -> LDS ----
    if (tid < Mdim) {
        float4 r0 = *(const float4*)(rho0 + ((size_t)n * Mdim + tid) * Cdim);
        *(float4*)&rho_lds[tid * Cdim] = r0;
    }

    // T[n] (uniform per block) into registers
    const float* Tn = T + (size_t)n * (Cdim * Cdim);
    float Tm[16];
    #pragma unroll
    for (int k = 0; k < 16; ++k) Tm[k] = Tn[k];

    __syncthreads();

    const v2f ones = {1.f, 1.f};

    for (int t = 0; t < timesteps; ++t) {
        // (a) record pre-update rho to trajectory
        if (tid < Mdim) {
            float4 r = *(const float4*)&rho_lds[tid * Cdim];
            *(float4*)(out + (((size_t)t * N + n) * Mdim + tid) * Cdim) = r;
        }

        // (b) s[i] = sum_j log2(1 - rho[i][1]*A[i][j])   (WMMA row-sum: L x ones)
        {
            const float rho1 = rho_lds[arow * Cdim + 1];
            v8f acc = {0.f, 0.f, 0.f, 0.f, 0.f, 0.f, 0.f, 0.f};
            #pragma unroll
            for (int kc = 0; kc < 16; ++kc) {
                const int kb = kc * 4 + hi * 2;
                v2f av = *(const v2f*)&A_lds[arow * APAD + kb];
                v2f a;
                a.x = hw_log2(__builtin_fmaf(-rho1, av.x, 1.f));
                a.y = hw_log2(__builtin_fmaf(-rho1, av.y, 1.f));
                acc = __builtin_amdgcn_wmma_f32_16x16x4_f32(
                        false, a, false, ones, (short)0, acc, false, false);
            }
            if ((lid & 15) == 0) {        // lanes 0 (M=0..7) and 16 (M=8..15)
                #pragma unroll
                for (int r = 0; r < 8; ++r)
                    s_lds[i0 + hi * 8 + r] = acc[r];
            }
        }
        __syncthreads();

        // (c) p = 1 - exp2(s)   (s is in log2 space)
        if (tid < Mdim) p_lds[tid] = 1.f - hw_exp2(s_lds[tid]);
        __syncthreads();

        // (d) q[i] = sum_k R[n,i,k] * p[k]   (WMMA: R x broadcast(p))
        {
            v8f acc = {0.f, 0.f, 0.f, 0.f, 0.f, 0.f, 0.f, 0.f};
            #pragma unroll
            for (int kc = 0; kc < 16; ++kc) {
                const int kb = kc * 4 + hi * 2;
                v2f a = *(const v2f*)&Rl[arow * APAD + kb];
                v2f b = *(const v2f*)&p_lds[kb];   // broadcast across 16 columns
                acc = __builtin_amdgcn_wmma_f32_16x16x4_f32(
                        false, a, false, b, (short)0, acc, false, false);
            }
            if ((lid & 15) == 0) {
                #pragma unroll
                for (int r = 0; r < 8; ++r)
                    q_lds[i0 + hi * 8 + r] = acc[r];
            }
        }
        __syncthreads();

        // (e,f,g) new_inf + rho@T + clip, write back
        if (tid < Mdim) {
            float4 r = *(const float4*)&rho_lds[tid * Cdim];
            const float tot  = r.x + r.y + r.z + r.w;
            const float ninf = (1.f - tot) * q_lds[tid];
            float o0 = r.x * Tm[0] + r.y * Tm[4] + r.z * Tm[8]  + r.w * Tm[12] + ninf;
            float o1 = r.x * Tm[1] + r.y * Tm[5] + r.z * Tm[9]  + r.w * Tm[13];
            float o2 = r.x * Tm[2] + r.y * Tm[6] + r.z * Tm[10] + r.w * Tm[14];
            float o3 = r.x * Tm[3] + r.y * Tm[7] + r.z * Tm[11] + r.w * Tm[15];
            float4 nr;
            nr.x = fminf(fmaxf(o0, 0.f), CLIP_MAX);
            nr.y = fminf(fmaxf(o1, 0.f), CLIP_MAX);
            nr.z = fminf(fmaxf(o2, 0.f), CLIP_MAX);
            nr.w = fminf(fmaxf(o3, 0.f), CLIP_MAX);
            *(float4*)&rho_lds[tid * Cdim] = nr;
        }
        __syncthreads();
    }
}

extern "C" void kernel_launch(void* const* d_in, const int* in_sizes, int n_in,
                              void* d_out, int out_size, void* d_ws, size_t ws_size,
                              hipStream_t stream) {
    (void)n_in; (void)d_ws; (void)ws_size;
    const float* R    = (const float*)d_in[0];
    const float* T    = (const float*)d_in[1];
    const float* rho0 = (const float*)d_in[2];
    const float* beta = (const float*)d_in[3];
    const int N = in_sizes[3];                          // beta has N elements
    const int timesteps = out_size / (N * Mdim * Cdim); // 100
    metapop_step_kernel<<<N, 128, 0, stream>>>(R, T, rho0, beta,
                                               (float*)d_out, N, timesteps);
}